// GCN_80376017977701
// MI455X (gfx1250) — compile-verified
//
#include <hip/hip_runtime.h>

typedef __attribute__((ext_vector_type(2))) float v2f;
typedef __attribute__((ext_vector_type(8))) float v8f;

#define NEG_SLOPE 0.01f

// ---------------------------------------------------------------------------
// Dense fp32 GEMM via V_WMMA_F32_16X16X4_F32.
// H[n_rows x D_OUT] = X[n_rows x 64] @ W[64 x D_OUT]   (K fixed at 64)
// One wave (32 threads) per 16x16 output tile => EXEC all ones for WMMA.
// grid = (n_rows/16, D_OUT/16), block = 32.
// D_OUT is a template constant so all strides fold into load immediates,
// eliminating the 64-bit address multiplies seen in the previous build.
// ---------------------------------------------------------------------------
template <int D_OUT>
__global__ void __launch_bounds__(32) gemm64_wmma(const float* __restrict__ X,
                                                  const float* __restrict__ W,
                                                  float* __restrict__ H) {
  const int lane  = (int)(threadIdx.x & 31u);
  const int row0  = (int)blockIdx.x << 4;
  const int col0  = (int)blockIdx.y << 4;
  const int m     = lane & 15;         // A: row M; B/C/D: column N
  const int khalf = (lane >> 4) << 1;  // lanes 0-15 -> K{0,1}; lanes 16-31 -> K{2,3}

  // Per-lane base pointers computed once; loop indices become immediates.
  const float* __restrict__ xp = X + (row0 + m) * 64 + khalf;
  const float* __restrict__ wp = W + khalf * D_OUT + col0 + m;

  v8f c = {};
#pragma unroll
  for (int kk = 0; kk < 16; ++kk) {  // 64 / 4 = 16 WMMA steps
    v2f a, b;
    a.x = xp[kk * 4];
    a.y = xp[kk * 4 + 1];
    b.x = wp[kk * 4 * D_OUT];
    b.y = wp[kk * 4 * D_OUT + D_OUT];
    // 8 args: (neg_a, A, neg_b, B, c_mod, C, reuse_a, reuse_b)
    c = __builtin_amdgcn_wmma_f32_16x16x4_f32(false, a, false, b, (short)0, c,
                                              false, false);
  }
  // C/D layout: VGPR r -> M = r (lanes 0-15) or r+8 (lanes 16-31), N = lane&15
  float* __restrict__ hp = H + (row0 + ((lane >> 4) << 3)) * D_OUT + col0 + m;
#pragma unroll
  for (int r = 0; r < 8; ++r) hp[r * D_OUT] = c[r];
}

// ---------------------------------------------------------------------------
// Degree / normalization precompute
// ---------------------------------------------------------------------------
__global__ void deg_init_k(float* __restrict__ deg, int n) {
  int i = blockIdx.x * blockDim.x + threadIdx.x;
  if (i < n) deg[i] = 1.0f;  // self-loop
}

__global__ void deg_count_k(const int* __restrict__ dst, float* __restrict__ deg,
                            int e) {
  int i = blockIdx.x * blockDim.x + threadIdx.x;
  if (i < e) atomicAdd(&deg[dst[i]], 1.0f);
}

__global__ void rsqrt_k(const float* __restrict__ deg, float* __restrict__ dis,
                        int n) {
  int i = blockIdx.x * blockDim.x + threadIdx.x;
  if (i < n) dis[i] = rsqrtf(deg[i]);
}

__global__ void norm_k(const int* __restrict__ src, const int* __restrict__ dst,
                       const float* __restrict__ dis, float* __restrict__ nrm,
                       int e) {
  int i = blockIdx.x * blockDim.x + threadIdx.x;
  if (i < e) nrm[i] = dis[src[i]] * dis[dst[i]];
}

// ---------------------------------------------------------------------------
// out[i,:] = b + h[i,:] * (1/deg[i])   (bias + self-loop contribution)
// grid = (ceil(n/256), D_OUT/4), block = 256; one float4 per thread.
// ---------------------------------------------------------------------------
template <int D_OUT>
__global__ void agg_init_k(const float* __restrict__ h,
                           const float* __restrict__ dis,
                           const float* __restrict__ bias,
                           float* __restrict__ out, int n) {
  int node = blockIdx.x * blockDim.x + threadIdx.x;
  if (node >= n) return;
  int g = (int)blockIdx.y << 2;
  float s = dis[node];
  s = s * s;  // 1/deg
  const float4 hv = *(const float4*)(h + node * D_OUT + g);
  const float4 bv = *(const float4*)(bias + g);
  float4 o;
  o.x = bv.x + hv.x * s;
  o.y = bv.y + hv.y * s;
  o.z = bv.z + hv.z * s;
  o.w = bv.w + hv.w * s;
  *(float4*)(out + node * D_OUT + g) = o;
}

// ---------------------------------------------------------------------------
// Edge scatter: out[dst[e],:] += h[src[e],:] * norm[e]
// grid = (ceil(E/256), D_OUT/4), block = 256; one float4 per thread.
// h/out are L2-resident (25.6 MB each << 192 MB L2) so gathers+atomics hit L2.
// ---------------------------------------------------------------------------
template <int D_OUT>
__global__ void agg_edges_k(const int* __restrict__ src,
                            const int* __restrict__ dst,
                            const float* __restrict__ nrm,
                            const float* __restrict__ h,
                            float* __restrict__ out, int e) {
  int ed = blockIdx.x * blockDim.x + threadIdx.x;
  if (ed >= e) return;
  int g = (int)blockIdx.y << 2;
  int s = src[ed];
  int d = dst[ed];
  float ne = nrm[ed];
  const float4 hv = *(const float4*)(h + s * D_OUT + g);
  float* op = out + d * D_OUT + g;
  atomicAdd(op + 0, hv.x * ne);
  atomicAdd(op + 1, hv.y * ne);
  atomicAdd(op + 2, hv.z * ne);
  atomicAdd(op + 3, hv.w * ne);
}

__global__ void leaky_k(float* __restrict__ x, int n) {
  int i = blockIdx.x * blockDim.x + threadIdx.x;
  if (i < n) {
    float v = x[i];
    x[i] = (v >= 0.0f) ? v : NEG_SLOPE * v;
  }
}

// ---------------------------------------------------------------------------
extern "C" void kernel_launch(void* const* d_in, const int* in_sizes, int n_in,
                              void* d_out, int out_size, void* d_ws,
                              size_t ws_size, hipStream_t stream) {
  (void)n_in;
  (void)out_size;
  (void)ws_size;

  const float* x  = (const float*)d_in[0];
  const int*   ei = (const int*)d_in[1];
  const float* W1 = (const float*)d_in[2];
  const float* b1 = (const float*)d_in[3];
  const float* W2 = (const float*)d_in[4];
  const float* b2 = (const float*)d_in[5];
  const float* W3 = (const float*)d_in[6];
  const float* b3 = (const float*)d_in[7];

  const int N = in_sizes[0] / 64;  // 100000
  const int E = in_sizes[1] / 2;   // 1600000
  const int* src = ei;
  const int* dst = ei + E;

  // workspace layout (floats): deg[N] | dis[N] | norm[E] | h[N*64]
  float* ws  = (float*)d_ws;
  float* deg = ws;
  float* dis = deg + N;
  float* nrm = dis + N;
  float* h   = nrm + E;

  float* emb1 = (float*)d_out;
  float* emb2 = emb1 + (size_t)N * 64;
  float* emb3 = emb2 + (size_t)N * 64;

  const int T = 256;
  const int nb_n = (N + T - 1) / T;
  const int nb_e = (E + T - 1) / T;
  const int nb_f = (N * 64 + T - 1) / T;

  // ---- normalization precompute (once, reused by all 3 layers) ----
  deg_init_k<<<nb_n, T, 0, stream>>>(deg, N);
  deg_count_k<<<nb_e, T, 0, stream>>>(dst, deg, E);
  rsqrt_k<<<nb_n, T, 0, stream>>>(deg, dis, N);
  norm_k<<<nb_e, T, 0, stream>>>(src, dst, dis, nrm, E);

  // ---- layer 1: 64 -> 64, leaky relu ----
  gemm64_wmma<64><<<dim3(N / 16, 4), 32, 0, stream>>>(x, W1, h);
  agg_init_k<64><<<dim3(nb_n, 16), T, 0, stream>>>(h, dis, b1, emb1, N);
  agg_edges_k<64><<<dim3(nb_e, 16), T, 0, stream>>>(src, dst, nrm, h, emb1, E);
  leaky_k<<<nb_f, T, 0, stream>>>(emb1, N * 64);

  // ---- layer 2: 64 -> 64, leaky relu ----
  gemm64_wmma<64><<<dim3(N / 16, 4), 32, 0, stream>>>(emb1, W2, h);
  agg_init_k<64><<<dim3(nb_n, 16), T, 0, stream>>>(h, dis, b2, emb2, N);
  agg_edges_k<64><<<dim3(nb_e, 16), T, 0, stream>>>(src, dst, nrm, h, emb2, E);
  leaky_k<<<nb_f, T, 0, stream>>>(emb2, N * 64);

  // ---- layer 3: 64 -> 32, no activation ----
  gemm64_wmma<32><<<dim3(N / 16, 2), 32, 0, stream>>>(emb2, W3, h);
  agg_init_k<32><<<dim3(nb_n, 8), T, 0, stream>>>(h, dis, b3, emb3, N);
  agg_edges_k<32><<<dim3(nb_e, 8), T, 0, stream>>>(src, dst, nrm, h, emb3, E);
}